// LandmarkGNN_4483945857186
// MI455X (gfx1250) — compile-verified
//
#include <hip/hip_runtime.h>
#include <hip/hip_bf16.h>

// Problem constants (match reference)
#define DN 50000
#define D_IN 128
#define D_H 256
#define D_OUT 128
#define N_MID 10
#define EPSV 1e-5f

typedef __attribute__((ext_vector_type(2))) float v2f;
typedef __attribute__((ext_vector_type(8))) float v8f;
typedef __attribute__((ext_vector_type(4))) unsigned int v4u;
typedef __attribute__((ext_vector_type(8))) int v8i;
typedef __attribute__((ext_vector_type(4))) int v4i;

#if defined(__HIP_DEVICE_COMPILE__)
#if __has_builtin(__builtin_amdgcn_tensor_load_to_lds) && __has_builtin(__builtin_amdgcn_s_wait_tensorcnt)
#define USE_TDM 1
#endif
#endif
#ifndef USE_TDM
#define USE_TDM 0
#endif

// ---------------- utility kernels ----------------

__global__ void fill_val_kernel(float* __restrict__ p, int n, float v) {
  int i = blockIdx.x * blockDim.x + threadIdx.x;
  if (i < n) p[i] = v;
}

// out[i] = bias[i % d], d is a power of two (128 or 256)
__global__ void fill_bias_kernel(float* __restrict__ out, const float* __restrict__ b,
                                 int n, int dmask) {
  int i = blockIdx.x * blockDim.x + threadIdx.x;
  if (i < n) out[i] = b[i & dmask];
}

// degree count with self loops
__global__ void deg_kernel(const int* __restrict__ dstE, float* __restrict__ deg,
                           int E, int EN) {
  int e = blockIdx.x * blockDim.x + threadIdx.x;
  if (e >= EN) return;
  int t = (e < E) ? dstE[e] : (e - E);
  unsafeAtomicAdd(&deg[t], 1.0f);
}

// deg -> d^{-1/2} in place
__global__ void dis_kernel(float* __restrict__ deg, int n) {
  int i = blockIdx.x * blockDim.x + threadIdx.x;
  if (i >= n) return;
  float d = deg[i];
  deg[i] = (d > 0.0f) ? rsqrtf(fmaxf(d, 1.0f)) : 0.0f;
}

__global__ void norm_kernel(const int* __restrict__ srcE, const int* __restrict__ dstE,
                            const float* __restrict__ dis, float* __restrict__ norm,
                            int E, int EN) {
  int e = blockIdx.x * blockDim.x + threadIdx.x;
  if (e >= EN) return;
  float v;
  if (e < E) {
    v = dis[srcE[e]] * dis[dstE[e]];
  } else {
    float z = dis[e - E];
    v = z * z;
  }
  norm[e] = v;
}

// ---------------- Tensor Data Mover helper (2D tile -> LDS) ----------------
// Builds D# group0/group1 per CDNA5 ISA:
//   group0: count=1 | lds_addr | global_addr(57b) | type=2
//   group1: data_size=4B, optional LDS padding, tensor dims/strides, 2D tile
// Toolchain here exposes the 6-arg builtin:
//   (uint32x4 g0, int32x8 g1, int32x4 g2, int32x4 g3, int32x8 extra, i32 cpol)
#if USE_TDM
__device__ __forceinline__ void tdm_load_2d(unsigned lds_addr, const void* gptr,
                                            unsigned dim0, unsigned dim1,
                                            unsigned stride0,
                                            unsigned tile0, unsigned tile1,
                                            unsigned pad_en, unsigned pad_interval) {
  unsigned long long ga = (unsigned long long)(uintptr_t)gptr;
  v4u g0;
  g0[0] = 1u;                                             // count=1, user desc
  g0[1] = lds_addr;                                       // LDS byte offset
  g0[2] = (unsigned)(ga & 0xFFFFFFFFull);                 // global_addr[31:0]
  g0[3] = (unsigned)((ga >> 32) & 0x1FFFFFFull) | (2u << 30); // addr[56:32] | type=2
  v8i g1;
  g1[0] = (int)((2u << 16) | (pad_en << 20) | (pad_interval << 22)); // data_size=4B
  g1[1] = (int)((dim0 & 0xFFFFu) << 16);                  // tensor_dim0[15:0]
  g1[2] = (int)((dim0 >> 16) | ((dim1 & 0xFFFFu) << 16)); // dim0[31:16] | dim1[15:0]
  g1[3] = (int)((dim1 >> 16) | (tile0 << 16));            // dim1[31:16] | tile_dim0
  g1[4] = (int)(tile1);                                   // tile_dim1 | tile_dim2=0
  g1[5] = (int)(stride0);                                 // tensor_dim0_stride[31:0]
  g1[6] = 0;                                              // stride0[47:32] | stride1 lo
  g1[7] = 0;
  v4i z4 = {0, 0, 0, 0};
  v8i z8 = {0, 0, 0, 0, 0, 0, 0, 0};
  __builtin_amdgcn_tensor_load_to_lds(g0, g1, z4, z4, z8, 0);
}
#endif

// ---------------- fp32 WMMA GEMM with LDS staging ----------------
// Block = 128 threads (4 waves). Block computes a 16-row x DOUT-col output slab.
// Per 32-wide K chunk: wave 0 DMAs (TDM) the A tile (16x32, padded rows: 33-float
// stride for bank-conflict-free column reads) and the W panel (32xDOUT) into LDS,
// waits TENSORcnt, barrier, then each wave runs K-steps of V_WMMA_F32_16X16X4_F32
// over its DOUT/4 columns (A fragment reused across DOUT/64 N-tiles).
//
// fp32 A-layout (16x4): lanes 0-15: M=lane, K={k0,k0+1}; lanes 16-31: K={k0+2,k0+3}.
// fp32 B-layout (4x16): VGPR0 rows k0/k0+2, VGPR1 rows k0+1/k0+3; N=lane&15.
// C/D (16x16): VGPR v: M=v (lanes 0-15) / v+8 (lanes 16-31), N=lane&15.
template <int DIN, int DOUT>
__global__ void gemm_wmma_tdm(const float* __restrict__ A, const float* __restrict__ W,
                              float* __restrict__ O) {
  constexpr int KC = 32;
  constexpr int APAD = KC + 1;     // 33-float row stride via TDM pad feature
  constexpr int TPW = DOUT / 64;   // 16x16 N-tiles per wave (4 waves)
  __shared__ float As[16 * APAD];
  __shared__ float Bs[KC * DOUT];

  const int wave = threadIdx.x >> 5;
  const int lane = threadIdx.x & 31;
  const int tileM = blockIdx.x;              // 50000/16 = 3125 exact
  const int nl = lane & 15;
  const int khalf = (lane >> 4) << 1;        // 0 (lanes 0-15) or 2 (lanes 16-31)

  v8f acc[TPW];
#pragma unroll
  for (int t = 0; t < TPW; ++t) acc[t] = (v8f){0.f,0.f,0.f,0.f,0.f,0.f,0.f,0.f};

  const float* aTile = A + (size_t)tileM * 16 * DIN;

  for (int k0 = 0; k0 < DIN; k0 += KC) {
#if USE_TDM
    if (wave == 0) {
      // A tile: 16 rows x KC cols; pad 1 DWORD every 32 DWORDs (interval code 4)
      tdm_load_2d((unsigned)(uintptr_t)&As[0], aTile + k0,
                  (unsigned)DIN, (unsigned)DN, (unsigned)DIN,
                  (unsigned)KC, 16u, 1u, 4u);
      // W panel: KC rows x DOUT cols, contiguous
      tdm_load_2d((unsigned)(uintptr_t)&Bs[0], W + (size_t)k0 * DOUT,
                  (unsigned)DOUT, (unsigned)DIN, (unsigned)DOUT,
                  (unsigned)DOUT, (unsigned)KC, 0u, 0u);
      __builtin_amdgcn_s_wait_tensorcnt(0);
    }
#else
    for (int idx = threadIdx.x; idx < 16 * KC; idx += 128) {
      int r = idx / KC, c = idx % KC;
      As[r * APAD + c] = aTile[(size_t)r * DIN + k0 + c];
    }
    for (int idx = threadIdx.x; idx < KC * DOUT; idx += 128) {
      int r = idx / DOUT, c = idx % DOUT;
      Bs[idx] = W[(size_t)(k0 + r) * DOUT + c];
    }
#endif
    __syncthreads();

#pragma unroll
    for (int kk = 0; kk < KC; kk += 4) {
      v2f a;
      a.x = As[nl * APAD + kk + khalf];
      a.y = As[nl * APAD + kk + khalf + 1];
#pragma unroll
      for (int t = 0; t < TPW; ++t) {
        const int n = (wave * TPW + t) * 16 + nl;
        v2f b;
        b.x = Bs[(kk + khalf) * DOUT + n];
        b.y = Bs[(kk + khalf + 1) * DOUT + n];
        acc[t] = __builtin_amdgcn_wmma_f32_16x16x4_f32(
            /*neg_a=*/false, a, /*neg_b=*/false, b,
            /*c_mod=*/(short)0, acc[t], /*reuse_a=*/false, /*reuse_b=*/false);
      }
    }
    __syncthreads();
  }

  const int rowBase = tileM * 16 + ((lane >> 4) << 3);
#pragma unroll
  for (int t = 0; t < TPW; ++t) {
    const int n = (wave * TPW + t) * 16 + nl;
#pragma unroll
    for (int v = 0; v < 8; ++v) {
      O[(size_t)(rowBase + v) * DOUT + n] = acc[t][v];
    }
  }
}

// ---------------- edge aggregation: out[dst] += lin[src] * norm (float4/thread)
__global__ void aggregate_kernel(const float* __restrict__ lin,
                                 const int* __restrict__ srcE, const int* __restrict__ dstE,
                                 const float* __restrict__ norm, float* __restrict__ out,
                                 int E, int EN, int d, int gshift) {
  unsigned tid = blockIdx.x * blockDim.x + threadIdx.x;
  unsigned total = (unsigned)EN << gshift;
  if (tid >= total) return;
  int e = tid >> gshift;
  int g = tid & ((1u << gshift) - 1u);
  int s, t;
  if (e < E) { s = srcE[e]; t = dstE[e]; }
  else       { s = t = e - E; }
  float nw = norm[e];
  const float4* lp = (const float4*)(lin + (size_t)s * d);
  float4 v = lp[g];
  float* op = out + (size_t)t * d + (g << 2);
  unsafeAtomicAdd(op + 0, v.x * nw);
  unsafeAtomicAdd(op + 1, v.y * nw);
  unsafeAtomicAdd(op + 2, v.z * nw);
  unsafeAtomicAdd(op + 3, v.w * nw);
}

// ---------------- BatchNorm: per-channel sum/sumsq partial reduction
__global__ void bn_stats_kernel(const float* __restrict__ h, float* __restrict__ stats,
                                int nrows, int d, int rowsPerBlock) {
  int c = threadIdx.x;
  int r0 = blockIdx.x * rowsPerBlock;
  int r1 = min(nrows, r0 + rowsPerBlock);
  float s = 0.f, s2 = 0.f;
  for (int r = r0; r < r1; ++r) {
    float v = h[(size_t)r * d + c];
    s += v;
    s2 += v * v;
  }
  unsafeAtomicAdd(&stats[c], s);
  unsafeAtomicAdd(&stats[d + c], s2);
}

__global__ void bn_relu_kernel(float* __restrict__ h, const float* __restrict__ stats,
                               const float* __restrict__ g, const float* __restrict__ be,
                               int n, int dmask, int d, float invN) {
  int i = blockIdx.x * blockDim.x + threadIdx.x;
  if (i >= n) return;
  int c = i & dmask;
  float mu = stats[c] * invN;
  float var = stats[d + c] * invN - mu * mu;
  float y = (h[i] - mu) * rsqrtf(var + EPSV) * g[c] + be[c];
  h[i] = y > 0.f ? y : 0.f;
}

// ---------------- orchestration ----------------

extern "C" void kernel_launch(void* const* d_in, const int* in_sizes, int n_in,
                              void* d_out, int out_size, void* d_ws, size_t ws_size,
                              hipStream_t stream) {
  const float* x   = (const float*)d_in[0];
  const int*   ei  = (const int*)d_in[1];
  const float* W1  = (const float*)d_in[2];
  const float* b1  = (const float*)d_in[3];
  const float* g1  = (const float*)d_in[4];
  const float* be1 = (const float*)d_in[5];
  const float* Wm  = (const float*)d_in[6];
  const float* bm  = (const float*)d_in[7];
  const float* gm  = (const float*)d_in[8];
  const float* bem = (const float*)d_in[9];
  const float* Wl  = (const float*)d_in[10];
  const float* bl  = (const float*)d_in[11];
  float* out = (float*)d_out;

  const int N  = in_sizes[0] / D_IN;   // 50000
  const int E  = in_sizes[1] / 2;      // 800000
  const int EN = E + N;
  const int* srcE = ei;
  const int* dstE = ei + E;

  // workspace layout (floats)
  float* P     = (float*)d_ws;                 // N x 256 (linear-transform output)
  float* Q     = P + (size_t)N * D_H;          // N x 256 (aggregated / activations)
  float* dis   = Q + (size_t)N * D_H;          // N
  float* norm  = dis + N;                      // E+N
  float* stats = norm + EN;                    // 2*256

  const int T = 256;
  const float invN = 1.0f / (float)N;

  // ---- normalization coefficients (once) ----
  fill_val_kernel<<<(N + T - 1) / T, T, 0, stream>>>(dis, N, 0.0f);
  deg_kernel<<<(EN + T - 1) / T, T, 0, stream>>>(dstE, dis, E, EN);
  dis_kernel<<<(N + T - 1) / T, T, 0, stream>>>(dis, N);
  norm_kernel<<<(EN + T - 1) / T, T, 0, stream>>>(srcE, dstE, dis, norm, E, EN);

  const int bnChunks = (N + 127) / 128;
  const int gemmBlocks = N / 16;               // 3125

  // ---- layer 1: 128 -> 256 ----
  {
    gemm_wmma_tdm<D_IN, D_H><<<gemmBlocks, 128, 0, stream>>>(x, W1, P);
    int n = N * D_H;
    fill_bias_kernel<<<(n + T - 1) / T, T, 0, stream>>>(Q, b1, n, D_H - 1);
    unsigned tot = (unsigned)EN << 6;
    aggregate_kernel<<<(tot + T - 1) / T, T, 0, stream>>>(P, srcE, dstE, norm, Q, E, EN, D_H, 6);
    fill_val_kernel<<<2, T, 0, stream>>>(stats, 2 * D_H, 0.0f);
    bn_stats_kernel<<<bnChunks, D_H, 0, stream>>>(Q, stats, N, D_H, 128);
    bn_relu_kernel<<<(n + T - 1) / T, T, 0, stream>>>(Q, stats, g1, be1, n, D_H - 1, D_H, invN);
  }

  // ---- middle layers: 256 -> 256, x10 ----
  for (int l = 0; l < N_MID; ++l) {
    const float* W  = Wm  + (size_t)l * D_H * D_H;
    const float* b  = bm  + (size_t)l * D_H;
    const float* gg = gm  + (size_t)l * D_H;
    const float* bb = bem + (size_t)l * D_H;
    gemm_wmma_tdm<D_H, D_H><<<gemmBlocks, 128, 0, stream>>>(Q, W, P);
    int n = N * D_H;
    fill_bias_kernel<<<(n + T - 1) / T, T, 0, stream>>>(Q, b, n, D_H - 1);
    unsigned tot = (unsigned)EN << 6;
    aggregate_kernel<<<(tot + T - 1) / T, T, 0, stream>>>(P, srcE, dstE, norm, Q, E, EN, D_H, 6);
    fill_val_kernel<<<2, T, 0, stream>>>(stats, 2 * D_H, 0.0f);
    bn_stats_kernel<<<bnChunks, D_H, 0, stream>>>(Q, stats, N, D_H, 128);
    bn_relu_kernel<<<(n + T - 1) / T, T, 0, stream>>>(Q, stats, gg, bb, n, D_H - 1, D_H, invN);
  }

  // ---- final layer: 256 -> 128, no BN/ReLU ----
  {
    gemm_wmma_tdm<D_H, D_OUT><<<gemmBlocks, 128, 0, stream>>>(Q, Wl, P);
    int n = N * D_OUT;
    fill_bias_kernel<<<(n + T - 1) / T, T, 0, stream>>>(out, bl, n, D_OUT - 1);
    unsigned tot = (unsigned)EN << 5;
    aggregate_kernel<<<(tot + T - 1) / T, T, 0, stream>>>(P, srcE, dstE, norm, out, E, EN, D_OUT, 5);
  }
}